// AffineTransformLayer_90159953478192
// MI455X (gfx1250) — compile-verified
//
#include <hip/hip_runtime.h>

// CDNA5 / gfx1250 spatial-transformer bilinear sampler.
//
//   Stage A (wave 0 per block): sampled = T(2x3) @ grid(3x16) via
//     V_WMMA_F32_16X16X4_F32  (A rows 0..1 = theta rows, K padded to 4;
//     B rows = [x; y; 1; 0] for the block's 16 pixel columns).
//     D row 0 = x coords, row 1 = y coords -> LDS broadcast.
//   Stage B (all 256 threads): 16 lanes x float4 per pixel -> each of the
//     4 neighbor gathers is one coalesced 256B line (global_load_b128),
//     bilinear blend in fp32, coalesced 256B store.
//
// Memory-bound: ~256MB out + L2-resident gathers (16MB/batch slice vs 192MB L2)
// => ~512MB HBM traffic => ~22us at 23.3 TB/s.

typedef __attribute__((ext_vector_type(2))) float v2f;
typedef __attribute__((ext_vector_type(8))) float v8f;

#define OUT_H 256
#define OUT_W 256
#define IN_H  256
#define IN_W  256
#define CCH   64

__global__ void __launch_bounds__(256)
affine_sampler_wmma(const float* __restrict__ X,
                    const float* __restrict__ theta,
                    float* __restrict__ out)
{
    constexpr int N = OUT_H * OUT_W;      // pixels per batch
    __shared__ float sx[16];
    __shared__ float sy[16];

    const int b      = blockIdx.y;        // batch
    const int base_n = blockIdx.x * 16;   // first output pixel of this tile
    const int tid    = threadIdx.x;

    // ---- Stage A: coordinates via WMMA (wave 0 only; EXEC all-ones) ----
    if (tid < 32) {
        const float* t = theta + b * 6;   // [i=0: t0 t1 t2 | i=1: t3 t4 t5]
        const int m  = tid & 15;          // A-matrix row (M)
        const int hi = tid >> 4;          // 0 -> K={0,1}, 1 -> K={2,3}

        // A: 16x4 fp32.  VGPR0: K=0 (lanes 0-15) / K=2 (lanes 16-31)
        //                VGPR1: K=1 (lanes 0-15) / K=3 (lanes 16-31)
        float ax = 0.0f, ay = 0.0f;
        if (m == 0)      { ax = hi ? t[2] : t[0]; ay = hi ? 0.0f : t[1]; }
        else if (m == 1) { ax = hi ? t[5] : t[3]; ay = hi ? 0.0f : t[4]; }
        v2f a; a.x = ax; a.y = ay;

        // B: 4x16 fp32, rows = [gx; gy; 1; 0], cols = 16 pixels of this tile.
        // VGPR0: K=0 (lanes 0-15) / K=2 (lanes 16-31)
        // VGPR1: K=1 (lanes 0-15) / K=3 (lanes 16-31)
        float bx, by;
        if (hi == 0) {
            const int n  = base_n + m;
            const int jo = n & (OUT_W - 1);
            const int io = n >> 8;
            bx = fmaf((float)jo, 2.0f / 255.0f, -1.0f);  // linspace(-1,1,256)
            by = fmaf((float)io, 2.0f / 255.0f, -1.0f);
        } else {
            bx = 1.0f;   // K=2 row: homogeneous ones
            by = 0.0f;   // K=3 row: zero padding
        }
        v2f bm; bm.x = bx; bm.y = by;

        v8f cz = {};
        // v_wmma_f32_16x16x4_f32: D = A x B + 0
        v8f d = __builtin_amdgcn_wmma_f32_16x16x4_f32(
            /*neg_a=*/false, a, /*neg_b=*/false, bm,
            /*c_mod=*/(short)0, cz, /*reuse_a=*/false, /*reuse_b=*/false);

        // D layout: VGPR v, lanes 0-15 -> M=v, N=lane. Row 0 = x, row 1 = y.
        if (hi == 0) { sx[m] = d[0]; sy[m] = d[1]; }
    }
    __syncthreads();

    // ---- Stage B: bilinear gather + blend (all 256 threads) ----
    const int p  = tid >> 4;   // pixel within tile (0..15)
    const int s  = tid & 15;   // channel slot (float4 -> channels 4s..4s+3)
    const int n  = base_n + p;
    const int co = s * 4;

    float x = sx[p];
    float y = sy[p];
    x = 0.5f * (x + 1.0f) * (float)IN_W;   // reference multiplies by W, not W-1
    y = 0.5f * (y + 1.0f) * (float)IN_H;

    int x0 = (int)x;                        // trunc toward zero == jnp astype
    int y0 = (int)y;
    int x1 = x0 + 1;
    int y1 = y0 + 1;
    x0 = x0 < 0 ? 0 : (x0 > IN_W - 1 ? IN_W - 1 : x0);
    x1 = x1 < 0 ? 0 : (x1 > IN_W - 1 ? IN_W - 1 : x1);
    y0 = y0 < 0 ? 0 : (y0 > IN_H - 1 ? IN_H - 1 : y0);
    y1 = y1 < 0 ? 0 : (y1 > IN_H - 1 ? IN_H - 1 : y1);

    const float x0f = (float)x0, x1f = (float)x1;
    const float y0f = (float)y0, y1f = (float)y1;
    const float wa = (x1f - x) * (y1f - y);
    const float wb = (x1f - x) * (y - y0f);
    const float wc = (x - x0f) * (y1f - y);
    const float wd = (x - x0f) * (y - y0f);

    const size_t baseB = (size_t)b * (size_t)(IN_H * IN_W) * CCH;

    const float4 pa = *(const float4*)(X + baseB + (size_t)(y0 * IN_W + x0) * CCH + co);
    const float4 pb = *(const float4*)(X + baseB + (size_t)(y1 * IN_W + x0) * CCH + co);
    const float4 pc = *(const float4*)(X + baseB + (size_t)(y0 * IN_W + x1) * CCH + co);
    const float4 pd = *(const float4*)(X + baseB + (size_t)(y1 * IN_W + x1) * CCH + co);

    float4 o;
    o.x = wa * pa.x + wb * pb.x + wc * pc.x + wd * pd.x;
    o.y = wa * pa.y + wb * pb.y + wc * pc.y + wd * pd.y;
    o.z = wa * pa.z + wb * pb.z + wc * pc.z + wd * pd.z;
    o.w = wa * pa.w + wb * pb.w + wc * pc.w + wd * pd.w;

    *(float4*)(out + baseB + (size_t)n * CCH + co) = o;
}

extern "C" void kernel_launch(void* const* d_in, const int* in_sizes, int n_in,
                              void* d_out, int out_size, void* d_ws, size_t ws_size,
                              hipStream_t stream)
{
    (void)in_sizes; (void)n_in; (void)out_size; (void)d_ws; (void)ws_size;

    const float* X     = (const float*)d_in[0];   // [16, 256, 256, 64] f32
    const float* theta = (const float*)d_in[1];   // [16, 6] f32
    float*       out   = (float*)d_out;           // [16, 256, 256, 64] f32

    // 4096 tiles of 16 pixels per batch, 16 batches.
    dim3 grid(4096, 16, 1);
    dim3 block(256, 1, 1);
    affine_sampler_wmma<<<grid, block, 0, stream>>>(X, theta, out);
}